// GCN_30726196036175
// MI455X (gfx1250) — compile-verified
//
#include <hip/hip_runtime.h>
#include <hip/hip_bf16.h>

typedef __attribute__((ext_vector_type(2))) float v2f;
typedef __attribute__((ext_vector_type(8))) float v8f;

#define NF 16  // hidden width H

// ---------------------------------------------------------------------------
// init: zero agg1/agg2, deg = 1.0 (self loop)
// ---------------------------------------------------------------------------
__global__ void gcn_init(float* __restrict__ deg, float* __restrict__ agg1,
                         float* __restrict__ agg2, int n) {
  int i = blockIdx.x * blockDim.x + threadIdx.x;
  int total = n * NF;
  if (i < total) { agg1[i] = 0.0f; agg2[i] = 0.0f; }
  if (i < n) deg[i] = 1.0f;  // self-loop contribution
}

// ---------------------------------------------------------------------------
// degree count over edges (dst side), then dinv = rsqrt(deg) in place
// ---------------------------------------------------------------------------
__global__ void gcn_deg(const long long* __restrict__ ei, float* __restrict__ deg, int E) {
  int e = blockIdx.x * blockDim.x + threadIdx.x;
  if (e >= E) return;
  int d = (int)ei[(size_t)E + e];
  atomicAdd(&deg[d], 1.0f);
}

__global__ void gcn_rsqrt(float* __restrict__ deg, int n) {
  int i = blockIdx.x * blockDim.x + threadIdx.x;
  if (i >= n) return;
  deg[i] = __frsqrt_rn(deg[i]);  // deg >= 1 always (self loop)
}

// ---------------------------------------------------------------------------
// GEMM1: h1[N,16] = x[N,128] @ W1[128,16]   via V_WMMA_F32_16X16X4_F32
// One wave handles one 16-row tile; K loop = 128/4 = 32 WMMAs.
// A 16x4 f32 layout: lane = M (lane&15); lane-half picks K {0,1} vs {2,3}.
// B 4x16 f32 layout: lane = N; lane-half picks K base 0 vs 2; vgpr adds 1.
// C/D 16x16 f32: 8 VGPRs, lane = N, vgpr v -> M = v + (lane>=16 ? 8 : 0).
// ---------------------------------------------------------------------------
__global__ void gcn_gemm1(const float* __restrict__ x, const float* __restrict__ W1,
                          float* __restrict__ h1, int nTiles) {
  int wave = blockIdx.x * (blockDim.x >> 5) + (threadIdx.x >> 5);
  if (wave >= nTiles) return;               // wave-uniform: EXEC stays all-1s
  int lane = threadIdx.x & 31;
  int m    = lane & 15;
  int koff = (lane >> 4) << 1;              // 0 or 2
  const float* xrow = x + (size_t)(wave * 16 + m) * 128;

  v8f c = {};
  #pragma unroll 4
  for (int k0 = 0; k0 < 128; k0 += 4) {
    v2f a, b;
    a.x = xrow[k0 + koff];
    a.y = xrow[k0 + koff + 1];
    b.x = W1[(k0 + koff)     * NF + m];
    b.y = W1[(k0 + koff + 1) * NF + m];
    c = __builtin_amdgcn_wmma_f32_16x16x4_f32(false, a, false, b,
                                              (short)0, c, false, false);
  }
  float* out = h1 + (size_t)(wave * 16) * NF;
  int mhi = (lane >> 4) * 8;
  #pragma unroll
  for (int v = 0; v < 8; ++v) out[(size_t)(v + mhi) * NF + m] = c[v];
}

// ---------------------------------------------------------------------------
// scatter-add of 16-float messages: agg[d] += feat[s] * dinv[s]*dinv[d]
// 1 thread / edge, 4x float4 gather (L2-resident: 6.4MB), 16 f32 atomics.
// ---------------------------------------------------------------------------
__global__ void gcn_scatter16(const long long* __restrict__ ei,
                              const float* __restrict__ feat,
                              const float* __restrict__ dinv,
                              float* __restrict__ agg, int E) {
  int e = blockIdx.x * blockDim.x + threadIdx.x;
  if (e >= E) return;
  int s = (int)ei[e];
  int d = (int)ei[(size_t)E + e];
  float w = dinv[s] * dinv[d];
  const float4* fs = (const float4*)(feat + (size_t)s * NF);
  float* ad = agg + (size_t)d * NF;
  #pragma unroll
  for (int q = 0; q < 4; ++q) {
    float4 v = fs[q];
    atomicAdd(&ad[q * 4 + 0], v.x * w);
    atomicAdd(&ad[q * 4 + 1], v.y * w);
    atomicAdd(&ad[q * 4 + 2], v.z * w);
    atomicAdd(&ad[q * 4 + 3], v.w * w);
  }
}

// ---------------------------------------------------------------------------
// layer-1 epilogue: z1 = relu(agg1 + h1*dinv^2 (self loop) + b1)
// ---------------------------------------------------------------------------
__global__ void gcn_self_relu(const float* __restrict__ agg1,
                              const float* __restrict__ h1,
                              const float* __restrict__ dinv,
                              const float* __restrict__ b1,
                              float* __restrict__ z1, int n) {
  int i = blockIdx.x * blockDim.x + threadIdx.x;
  if (i >= n * NF) return;
  int node = i >> 4, j = i & 15;
  float di = dinv[node];
  float v = agg1[i] + h1[i] * di * di + b1[j];
  z1[i] = fmaxf(v, 0.0f);
}

// ---------------------------------------------------------------------------
// GEMM2 (+ fused self-loop, bias, ReLU):
//   out[N,128] = relu( (agg2 + z1*dinv^2)[N,16] @ W2[16,128] + b2 )
// One wave per 16x16 output tile; 8 column tiles per 16-row block; K=16 -> 4 WMMAs.
// ---------------------------------------------------------------------------
__global__ void gcn_gemm2(const float* __restrict__ agg2, const float* __restrict__ z1,
                          const float* __restrict__ dinv, const float* __restrict__ W2,
                          const float* __restrict__ b2, float* __restrict__ out,
                          int nWaves) {
  int wave = blockIdx.x * (blockDim.x >> 5) + (threadIdx.x >> 5);
  if (wave >= nWaves) return;               // wave-uniform
  int rowTile = wave >> 3;
  int nTile   = wave & 7;
  int colBase = nTile * 16;
  int lane = threadIdx.x & 31;
  int m    = lane & 15;
  int koff = (lane >> 4) << 1;
  int row  = rowTile * 16 + m;

  float di  = dinv[row];
  float di2 = di * di;
  const float* ar = agg2 + (size_t)row * NF;
  const float* zr = z1   + (size_t)row * NF;

  v8f c = {};
  #pragma unroll
  for (int k0 = 0; k0 < NF; k0 += 4) {
    v2f a, b;
    a.x = ar[k0 + koff]     + zr[k0 + koff]     * di2;
    a.y = ar[k0 + koff + 1] + zr[k0 + koff + 1] * di2;
    b.x = W2[(k0 + koff)     * 128 + colBase + m];
    b.y = W2[(k0 + koff + 1) * 128 + colBase + m];
    c = __builtin_amdgcn_wmma_f32_16x16x4_f32(false, a, false, b,
                                              (short)0, c, false, false);
  }
  int mhi = (lane >> 4) * 8;
  float bias = b2[colBase + m];
  #pragma unroll
  for (int v = 0; v < 8; ++v) {
    float val = c[v] + bias;
    out[(size_t)(rowTile * 16 + v + mhi) * 128 + colBase + m] = fmaxf(val, 0.0f);
  }
}

// ---------------------------------------------------------------------------
extern "C" void kernel_launch(void* const* d_in, const int* in_sizes, int n_in,
                              void* d_out, int out_size, void* d_ws, size_t ws_size,
                              hipStream_t stream) {
  const float*     x  = (const float*)d_in[0];
  const long long* ei = (const long long*)d_in[1];  // int64 [2,E]
  const float*     W1 = (const float*)d_in[2];
  const float*     b1 = (const float*)d_in[3];
  const float*     W2 = (const float*)d_in[4];
  const float*     b2 = (const float*)d_in[5];
  float*           out = (float*)d_out;

  const int N = in_sizes[0] / 128;   // 100000
  const int E = in_sizes[1] / 2;     // 1600000

  // workspace layout (floats): dinv | h1 | agg1 | z1 | agg2  = N*(1+64) floats
  float* ws   = (float*)d_ws;
  float* dinv = ws;
  float* h1   = dinv + (size_t)N;
  float* agg1 = h1   + (size_t)N * NF;
  float* z1   = agg1 + (size_t)N * NF;
  float* agg2 = z1   + (size_t)N * NF;

  const int T = 256;
  int gNF    = (N * NF + T - 1) / T;   // 6250
  int gE     = (E + T - 1) / T;        // 6250
  int gN     = (N + T - 1) / T;
  int tiles1 = (N + 15) / 16;          // 6250 waves
  int g1     = (tiles1 + 7) / 8;       // 8 waves per 256-thread block
  int waves2 = tiles1 * 8;             // 50000
  int g2     = (waves2 + 7) / 8;       // 6250

  gcn_init     <<<gNF, T, 0, stream>>>(dinv, agg1, agg2, N);
  gcn_deg      <<<gE,  T, 0, stream>>>(ei, dinv, E);
  gcn_rsqrt    <<<gN,  T, 0, stream>>>(dinv, N);
  gcn_gemm1    <<<g1,  T, 0, stream>>>(x, W1, h1, tiles1);
  gcn_scatter16<<<gE,  T, 0, stream>>>(ei, h1, dinv, agg1, E);
  gcn_self_relu<<<gNF, T, 0, stream>>>(agg1, h1, dinv, b1, z1, N);
  gcn_scatter16<<<gE,  T, 0, stream>>>(ei, z1, dinv, agg2, E);
  gcn_gemm2    <<<g2,  T, 0, stream>>>(agg2, z1, dinv, W2, b2, out, waves2);
}